// LinearSGC1_9594956939362
// MI455X (gfx1250) — compile-verified
//
#include <hip/hip_runtime.h>
#include <hip/hip_bf16.h>

#define D_FEAT 64
#define ROW_BYTES (D_FEAT * 4)      // 256 B per feature row
#define ROWS_PER_TDM 8              // max rows per gather descriptor, 32-bit index mode
#define EPW 32                      // edges per wave (4 TDM batches of 8)
#define WAVES_PER_BLOCK 8           // 256 threads

typedef unsigned int v4u __attribute__((ext_vector_type(4)));
typedef int          v4i __attribute__((ext_vector_type(4)));
typedef int          v8i __attribute__((ext_vector_type(8)));

// Native fp32 global atomic add (GLOBAL_ATOMIC_ADD_F32, L2-side RMW).
__device__ __forceinline__ void atomic_add_f32(float* p, float v) {
    __hip_atomic_fetch_add(p, v, __ATOMIC_RELAXED, __HIP_MEMORY_SCOPE_AGENT);
}

// deg[i] = 1.0f : the self-loop contributes 1 to every node's degree.
__global__ void k_init_deg(float* __restrict__ deg, int n) {
    int i = blockIdx.x * blockDim.x + threadIdx.x;
    if (i < n) deg[i] = 1.0f;
}

// deg[col[e]] += 1 over the real edges.
__global__ void k_count_deg(const int* __restrict__ col, float* __restrict__ deg, int E) {
    int e = blockIdx.x * blockDim.x + threadIdx.x;
    if (e < E) {
        int c = __builtin_nontemporal_load(col + e);
        atomic_add_f32(deg + c, 1.0f);
    }
}

// dis[i] = rsqrt(deg[i]); out[i,:] = dis[i]^2 * x[i,:]  (self-loop term; also
// initializes the poisoned output). 16 threads/node, float4 -> b128 ld/st.
__global__ void k_finalize(const float* __restrict__ x, const float* __restrict__ deg,
                           float* __restrict__ dis, float* __restrict__ out, int n) {
    int t = blockIdx.x * blockDim.x + threadIdx.x;
    int node = t >> 4;
    if (node >= n) return;
    int q = t & 15;
    float d  = deg[node];
    float di = d > 0.0f ? rsqrtf(fmaxf(d, 1.0f)) : 0.0f;
    if (q == 0) dis[node] = di;
    float s = di * di;
    size_t off = (size_t)node * D_FEAT + (size_t)q * 4;
    float4 xv = *(const float4*)(x + off);
    *(float4*)(out + off) = make_float4(s * xv.x, s * xv.y, s * xv.z, s * xv.w);
}

// out[row,:] += dis[row]*dis[col] * x[col,:]
// Each wave32 handles 32 edges. The x[col] row gathers (8 rows = 2KB per
// descriptor) are done by the Tensor Data Mover in gather mode (32-bit row
// indices in D# groups 2/3), double-buffered in LDS and drained with
// s_wait_tensorcnt, overlapping the DMA gather with the fp32 atomic scatter.
__global__ void __launch_bounds__(WAVES_PER_BLOCK * 32)
k_edge_scatter_tdm(const int* __restrict__ ei, const float* __restrict__ dis,
                   const float* __restrict__ x, float* __restrict__ out,
                   int E, int n) {
    __shared__ float stage[WAVES_PER_BLOCK][2][ROWS_PER_TDM][D_FEAT]; // 32 KB

    const int lane  = threadIdx.x & 31;
    const int wvloc = threadIdx.x >> 5;
    const long long wave = (long long)blockIdx.x * WAVES_PER_BLOCK + wvloc;
    const long long base = wave * EPW;
    if (base >= E) return;

    // Per-lane edge data for this wave's 32 edges (coalesced index streams).
    int r = 0, c = 0;
    float nrm = 0.0f;
    long long e = base + lane;
    if (e < E) {
        r = __builtin_nontemporal_load(ei + e);
        c = __builtin_nontemporal_load(ei + E + e);
        nrm = dis[r] * dis[c];
    }

    if (base + EPW <= E) {
        // LDS byte offset of this wave's staging area: flat LDS pointers on
        // gfx1250 carry the LDS offset in their low 32 bits.
        const unsigned lds0 = (unsigned)(size_t)&stage[wvloc][0][0][0];
        const unsigned long long xa = (unsigned long long)(size_t)x;

        // D# group 1 (shared by all batches):
        //  data_size=4B, tensor_dim0=64, tensor_dim1=n (OOB bound),
        //  tile_dim0=64, tile_dim1=8 valid indices, dim0_stride=64.
        v8i g1;
        g1[0] = 0x00020000;                                  // data_size code 2
        g1[1] = (D_FEAT & 0xFFFF) << 16;                     // tensor_dim0[15:0]
        g1[2] = (n & 0xFFFF) << 16;                          // tensor_dim1[15:0] | dim0 hi=0
        g1[3] = (D_FEAT << 16) | ((n >> 16) & 0xFFFF);       // tile_dim0 | tensor_dim1 hi
        g1[4] = ROWS_PER_TDM;                                // tile_dim1 = #indices
        g1[5] = D_FEAT;                                      // tensor_dim0_stride lo32
        g1[6] = 0;
        g1[7] = 0;

        auto issue = [&](int b) {
            const int l0 = b * ROWS_PER_TDM;
            v4i g2, g3;                                      // 32-bit row indices
            g2[0] = __builtin_amdgcn_readlane(c, l0 + 0);
            g2[1] = __builtin_amdgcn_readlane(c, l0 + 1);
            g2[2] = __builtin_amdgcn_readlane(c, l0 + 2);
            g2[3] = __builtin_amdgcn_readlane(c, l0 + 3);
            g3[0] = __builtin_amdgcn_readlane(c, l0 + 4);
            g3[1] = __builtin_amdgcn_readlane(c, l0 + 5);
            g3[2] = __builtin_amdgcn_readlane(c, l0 + 6);
            g3[3] = __builtin_amdgcn_readlane(c, l0 + 7);
            v4u g0;
            g0[0] = 0xC0000001u;            // count=1 | gather_index_size=32b | gather_mode
            g0[1] = lds0 + (unsigned)(b & 1) * (ROWS_PER_TDM * ROW_BYTES);
            g0[2] = (unsigned)(xa & 0xFFFFFFFFu);            // global_addr[31:0]
            g0[3] = (unsigned)((xa >> 32) & 0x01FFFFFFu)     // global_addr[56:32]
                  | 0x80000000u;                             // type=2 ("image")
            asm volatile("tensor_load_to_lds %0, %1, %2, %3"
                         :: "s"(g0), "s"(g1), "s"(g2), "s"(g3)
                         : "memory");
        };

        const int NB = EPW / ROWS_PER_TDM;   // 4 batches, ping-pong buffers
        issue(0);
#pragma unroll
        for (int b = 0; b < NB; ++b) {
            if (b + 1 < NB) {
                issue(b + 1);                                // prefetch next batch
                __builtin_amdgcn_s_wait_tensorcnt(1);        // batch b landed
            } else {
                __builtin_amdgcn_s_wait_tensorcnt(0);
            }
#pragma unroll
            for (int j = 0; j < ROWS_PER_TDM; ++j) {
                const int idx = b * ROWS_PER_TDM + j;
                const int rr = __builtin_amdgcn_readlane(r, idx);
                const float nn = __int_as_float(
                    __builtin_amdgcn_readlane(__float_as_int(nrm), idx));
                const float2 v =
                    *(const float2*)&stage[wvloc][b & 1][j][lane * 2]; // ds_load_b64
                float* o = out + (size_t)rr * D_FEAT + lane * 2;
                atomic_add_f32(o + 0, nn * v.x);
                atomic_add_f32(o + 1, nn * v.y);
            }
        }
    } else if (e < E) {
        // Generic tail (never taken for E % EPW == 0): per-lane direct scatter.
        const float* xr = x + (size_t)c * D_FEAT;
        float* o = out + (size_t)r * D_FEAT;
        for (int f = 0; f < D_FEAT; ++f)
            atomic_add_f32(o + f, nrm * xr[f]);
    }
}

extern "C" void kernel_launch(void* const* d_in, const int* in_sizes, int n_in,
                              void* d_out, int out_size, void* d_ws, size_t ws_size,
                              hipStream_t stream) {
    const float* x   = (const float*)d_in[0];   // [N, 64] fp32
    const int*   ei  = (const int*)d_in[1];     // [2, E]: rows = ei[0:E], cols = ei[E:2E]
    float*       out = (float*)d_out;           // [N, 64] fp32

    const int n = in_sizes[0] / D_FEAT;
    const int E = in_sizes[1] / 2;

    float* deg = (float*)d_ws;   // N floats
    float* dis = deg + n;        // N floats (800 KB total)

    const int B = 256;
    k_init_deg <<<(n + B - 1) / B,        B, 0, stream>>>(deg, n);
    k_count_deg<<<(E + B - 1) / B,        B, 0, stream>>>(ei + E, deg, E);
    k_finalize <<<((n * 16) + B - 1) / B, B, 0, stream>>>(x, deg, dis, out, n);

    // 8 waves/block * 32 edges/wave = 256 edges per block
    const int edges_per_block = WAVES_PER_BLOCK * EPW;
    k_edge_scatter_tdm<<<(E + edges_per_block - 1) / edges_per_block, B, 0, stream>>>(
        ei, dis, x, out, E, n);
}